// PureCartesianICTDO3TransformerLayer_84344567759198
// MI455X (gfx1250) — compile-verified
//
#include <hip/hip_runtime.h>

typedef __attribute__((ext_vector_type(2))) float v2f;
typedef __attribute__((ext_vector_type(4))) float v4f;
typedef __attribute__((ext_vector_type(8))) float v8f;
typedef __attribute__((ext_vector_type(4))) int   v4i;

#define AS1 __attribute__((address_space(1)))
#define AS3 __attribute__((address_space(3)))

#define CCH   128      // channels per irrep
#define DTOT  2304     // full feature width
#define WSTR  132      // LDS stride for W (128 + 4 pad; 16B aligned, conflict-free)
#define TNODE 32       // nodes per workgroup (100000 = 32 * 3125, no tail)

#if defined(__gfx1250__) && __has_builtin(__builtin_amdgcn_global_load_async_to_lds_b128)
#define ASYNC_LDS 1
#else
#define ASYNC_LDS 0
#endif

// Stage 16 bytes global -> LDS. Async (ASYNCcnt-tracked, no VGPR round trip) when available.
// Builtin prototype (LLVM): void(v4i AS1* src, v4i AS3* dst, constant int offset, constant int cpol)
__device__ __forceinline__ void stage16(float* lds_dst, const float* g_src) {
#if ASYNC_LDS
    __builtin_amdgcn_global_load_async_to_lds_b128(
        (AS1 v4i*)g_src,
        (AS3 v4i*)lds_dst,
        0, 0);
#else
    *(v4f*)lds_dst = *(const v4f*)g_src;
#endif
}

__device__ __forceinline__ void wait_stage() {
#if ASYNC_LDS
#if __has_builtin(__builtin_amdgcn_s_wait_asynccnt)
    __builtin_amdgcn_s_wait_asynccnt(0);
#else
    asm volatile("s_wait_asynccnt 0" ::: "memory");
#endif
#endif
}

// One irrep block: Y[n, d, mm] = sum_c W[d,c] * X[n, base + c*M + mm] ; out = x + Y
// M = 2l+1 (compile-time). Block = 256 threads = 8 wave32.
template<int M>
__device__ __forceinline__ void irrep_body(const float* __restrict__ x,
                                           const float* __restrict__ W,
                                           float* __restrict__ out,
                                           int base) {
    extern __shared__ float lds[];
    float* sW = lds;                        // 128 x WSTR
    float* sX = lds + CCH * WSTR;           // TNODE x XSTR
    constexpr int XSTR = CCH * M + 4;       // padded row stride (floats), 16B aligned
    constexpr int ROW4 = (CCH * M) / 4;     // float4s per data row

    const int tid   = threadIdx.x;
    const int node0 = blockIdx.x * TNODE;

    // ---- stage W[128][128] -> sW (stride WSTR), coalesced 16B, async ----
    #pragma unroll
    for (int i = tid; i < CCH * CCH / 4; i += 256) {
        int row  = i >> 5;          // 32 float4s per row
        int col4 = i & 31;
        stage16(&sW[row * WSTR + col4 * 4], &W[row * CCH + col4 * 4]);
    }
    // ---- stage X slab: TNODE rows x (128*M) floats -> sX (stride XSTR), async ----
    for (int i = tid; i < TNODE * ROW4; i += 256) {
        int row = i / ROW4;
        int c4  = i % ROW4;
        stage16(&sX[row * XSTR + c4 * 4],
                &x[(size_t)(node0 + row) * DTOT + base + c4 * 4]);
    }
    wait_stage();
    __syncthreads();

    const int wave  = tid >> 5;      // 0..7 (wave32)
    const int lane  = tid & 31;
    const int lr    = lane & 15;     // lane row 0..15
    const int lh    = lane >> 4;     // lane half 0/1
    const int rhalf = wave >> 2;     // node half: rows rhalf*16 .. +15
    const int d0    = (wave & 3) * 32;   // 32 output channels: tiles d0, d0+16

    // Accumulators: M components x 2 output tiles (<= 80 VGPRs for M=5)
    v8f acc0[M], acc1[M];
    #pragma unroll
    for (int mm = 0; mm < M; ++mm) { acc0[mm] = (v8f){}; acc1[mm] = (v8f){}; }

    // A-row base for this lane: X[node0 + rhalf*16 + lr, base + c*M + mm]
    const float* sXr = sX + (rhalf * 16 + lr) * XSTR;
    const float* sW0 = sW + (d0 +      lr) * WSTR + 2 * lh;
    const float* sW1 = sW + (d0 + 16 + lr) * WSTR + 2 * lh;

    // K-outer / mm-inner: each B fragment loaded once, reused across all M components.
    #pragma unroll 2
    for (int c = 0; c < CCH; c += 4) {
        // B (4x16): B[k][n] = W[d0+n][c+k]; VGPR j, lane half lh -> k = 2*lh + j
        v2f b0 = { sW0[c], sW0[c + 1] };
        v2f b1 = { sW1[c], sW1[c + 1] };
        #pragma unroll
        for (int mm = 0; mm < M; ++mm) {
            // A (16x4 f32): VGPR j, lane half lh -> K = c + 2*lh + j
            v2f a = { sXr[(c + 2 * lh)     * M + mm],
                      sXr[(c + 2 * lh + 1) * M + mm] };
            acc0[mm] = __builtin_amdgcn_wmma_f32_16x16x4_f32(false, a, false, b0,
                                                             (short)0, acc0[mm], false, false);
            acc1[mm] = __builtin_amdgcn_wmma_f32_16x16x4_f32(false, a, false, b1,
                                                             (short)0, acc1[mm], false, false);
        }
    }

    // All A-fragment reads of sX are done; fuse residual in place (each element
    // has exactly one owner lane), then copy out coalesced.
    __syncthreads();

    #pragma unroll
    for (int mm = 0; mm < M; ++mm) {
        #pragma unroll
        for (int v = 0; v < 8; ++v) {
            const int row = rhalf * 16 + v + 8 * lh;   // D: VGPR v -> row v + 8*lh
            const int da  = d0 + lr;
            const int db  = d0 + 16 + lr;
            sX[row * XSTR + da * M + mm] += acc0[mm][v];
            sX[row * XSTR + db * M + mm] += acc1[mm][v];
        }
    }

    __syncthreads();

    // ---- coalesced copy-out: full 16B lanes, full cachelines ----
    for (int i = tid; i < TNODE * ROW4; i += 256) {
        int row = i / ROW4;
        int c4  = i % ROW4;
        *(v4f*)&out[(size_t)(node0 + row) * DTOT + base + c4 * 4] =
            *(const v4f*)&sX[row * XSTR + c4 * 4];
    }
}

// Single launch: gridDim.y = 6 selects the irrep block (uniform branch per
// workgroup, EXEC stays all-ones around WMMA). Lets the scheduler interleave
// all 18750 workgroups and removes inter-kernel serialization gaps.
__launch_bounds__(256)
__global__ void irrep_mix_all(const float* __restrict__ x,
                              const float* __restrict__ W0p,
                              const float* __restrict__ W0m,
                              const float* __restrict__ W1p,
                              const float* __restrict__ W1m,
                              const float* __restrict__ W2p,
                              const float* __restrict__ W2m,
                              float* __restrict__ out) {
    switch (blockIdx.y) {
        case 0:  irrep_body<1>(x, W0p, out, 0);    break;
        case 1:  irrep_body<1>(x, W0m, out, 128);  break;
        case 2:  irrep_body<3>(x, W1p, out, 256);  break;
        case 3:  irrep_body<3>(x, W1m, out, 640);  break;
        case 4:  irrep_body<5>(x, W2p, out, 1024); break;
        default: irrep_body<5>(x, W2m, out, 1664); break;
    }
}

extern "C" void kernel_launch(void* const* d_in, const int* in_sizes, int n_in,
                              void* d_out, int out_size, void* d_ws, size_t ws_size,
                              hipStream_t stream) {
    const float* x   = (const float*)d_in[0];
    const float* W0p = (const float*)d_in[1];
    const float* W0m = (const float*)d_in[2];
    const float* W1p = (const float*)d_in[3];
    const float* W1m = (const float*)d_in[4];
    const float* W2p = (const float*)d_in[5];
    const float* W2m = (const float*)d_in[6];
    float* out = (float*)d_out;

    const int ntiles = 100000 / TNODE;   // 3125
    dim3 grid(ntiles, 6), block(256);

    // LDS for the largest (M=5) path: 128*132*4 + 32*(128*5+4)*4 = 150016 B
    const size_t lds_max = (size_t)CCH * WSTR * sizeof(float)
                         + (size_t)TNODE * (CCH * 5 + 4) * sizeof(float);

    irrep_mix_all<<<grid, block, lds_max, stream>>>(x, W0p, W0m, W1p, W1m, W2p, W2m, out);
}